// DecoderWithAttention_88759794139675
// MI455X (gfx1250) — compile-verified
//
#include <hip/hip_runtime.h>
#include <math.h>

// Shapes from the reference
#define NB 128      // batch
#define NP 196      // pixels
#define NE 2048     // encoder dim
#define ND 512      // decoder/attention/embedding dim
#define NV 18022    // vocab
#define NT 51       // decode steps (MAXLEN-1)
#define MAXLEN 52

typedef float v2f __attribute__((ext_vector_type(2)));
typedef float v8f __attribute__((ext_vector_type(8)));

__device__ __forceinline__ float sigmoidf_(float x) { return 1.0f / (1.0f + expf(-x)); }

// ---------------------------------------------------------------------------
// WMMA f32 16x16x4 (codegen-confirmed on gfx1250).
// VGPR layout (ISA 7.12.2, 32-bit):
//   A (16x4): v[j], lanes 0-15: A[m=lane][k0+j]; lanes 16-31: A[m=lane-16][k0+2+j]
//   B (4x16): v[j], lanes 0-15: B[k0+j][n=lane]; lanes 16-31: B[k0+2+j][n=lane-16]
//   C/D     : v[i], lanes 0-15: C[m=i][n=lane];  lanes 16-31: C[m=i+8][n=lane-16]
// ---------------------------------------------------------------------------
__device__ __forceinline__ v8f wmma4(v2f a, v2f b, v8f acc) {
  return __builtin_amdgcn_wmma_f32_16x16x4_f32(false, a, false, b, (short)0, acc,
                                               false, false);
}

// ---------------------------------------------------------------------------
// Generic fp32 WMMA GEMM:  C = A(MxK) * B(KxN) [+ A2(MxK2) * B2(K2xN)] + bias1 + bias2
// Block = 256 threads = 8 waves arranged 2 (M) x 4 (N); each wave computes a
// 32x32 tile (4 accumulators, 4 wmma per K-step). Block tile: 64 x 128.
// Requires M % 64 == 0, K % 4 == 0, K2 % 4 == 0. N may be ragged: out-of-range
// columns are loaded with a clamped index (garbage confined to lanes whose
// stores are guarded) so the hot loop has NO exec-mask manipulation.
// If act_len != null: row m is batch index; rows with !(t < act_len[m]-1) store 0.
// ---------------------------------------------------------------------------
__global__ __launch_bounds__(256) void gemm_wmma_f32(
    const float* __restrict__ A, int lda, const float* __restrict__ B, int ldb,
    const float* __restrict__ A2, int lda2, const float* __restrict__ B2, int ldb2,
    const float* __restrict__ bias1, const float* __restrict__ bias2,
    float* __restrict__ C, int ldc, int N, int K, int K2,
    const int* __restrict__ act_len, int t) {
  const int wave = threadIdx.x >> 5;
  const int lane = threadIdx.x & 31;
  const int half = lane >> 4;       // 0: lanes 0-15, 1: lanes 16-31
  const int lm = lane & 15;
  const int halfoff = half ? 2 : 0; // K offset for this half of the wave
  const int m0 = blockIdx.y * 64 + (wave >> 2) * 32;
  const int n0 = blockIdx.x * 128 + (wave & 3) * 32;

  const int na = n0 + lm;        // n for subtile 0 in this lane
  const int nb = n0 + 16 + lm;   // n for subtile 1 in this lane
  const int nc0 = (na < N) ? na : (N - 1);   // clamped (always-valid) columns
  const int nc1 = (nb < N) ? nb : (N - 1);

  v8f acc00 = {}, acc01 = {}, acc10 = {}, acc11 = {};

  {
    const float* Ap0 = A + (size_t)(m0 + lm) * lda + halfoff;
    const float* Ap1 = Ap0 + (size_t)16 * lda;
    const float* Bp0 = B + (size_t)halfoff * ldb + nc0;
    const float* Bp1 = B + (size_t)halfoff * ldb + nc1;
    const size_t bstep = (size_t)4 * ldb;
#pragma unroll 4
    for (int k0 = 0; k0 < K; k0 += 4) {
      v2f a0 = *(const v2f*)Ap0;
      v2f a1 = *(const v2f*)Ap1;
      v2f b0, b1;
      b0.x = Bp0[0]; b0.y = Bp0[ldb];
      b1.x = Bp1[0]; b1.y = Bp1[ldb];
      acc00 = wmma4(a0, b0, acc00);
      acc01 = wmma4(a0, b1, acc01);
      acc10 = wmma4(a1, b0, acc10);
      acc11 = wmma4(a1, b1, acc11);
      Ap0 += 4; Ap1 += 4; Bp0 += bstep; Bp1 += bstep;
    }
  }

  if (A2 != nullptr) {
    const float* Ap0 = A2 + (size_t)(m0 + lm) * lda2 + halfoff;
    const float* Ap1 = Ap0 + (size_t)16 * lda2;
    const float* Bp0 = B2 + (size_t)halfoff * ldb2 + nc0;
    const float* Bp1 = B2 + (size_t)halfoff * ldb2 + nc1;
    const size_t bstep = (size_t)4 * ldb2;
#pragma unroll 4
    for (int k0 = 0; k0 < K2; k0 += 4) {
      v2f a0 = *(const v2f*)Ap0;
      v2f a1 = *(const v2f*)Ap1;
      v2f b0, b1;
      b0.x = Bp0[0]; b0.y = Bp0[ldb2];
      b1.x = Bp1[0]; b1.y = Bp1[ldb2];
      acc00 = wmma4(a0, b0, acc00);
      acc01 = wmma4(a0, b1, acc01);
      acc10 = wmma4(a1, b0, acc10);
      acc11 = wmma4(a1, b1, acc11);
      Ap0 += 4; Ap1 += 4; Bp0 += bstep; Bp1 += bstep;
    }
  }

  // Epilogue: bias, activity masking, guarded stores.
  const bool in0 = (na < N);
  const bool in1 = (nb < N);
  const float bb0 = (bias1 ? bias1[nc0] : 0.0f) + (bias2 ? bias2[nc0] : 0.0f);
  const float bb1 = (bias1 ? bias1[nc1] : 0.0f) + (bias2 ? bias2[nc1] : 0.0f);
#pragma unroll
  for (int i = 0; i < 8; ++i) {
    const int mA = m0 + i + (half ? 8 : 0);
    const int mB = mA + 16;
    float v00 = acc00[i] + bb0, v01 = acc01[i] + bb1;
    float v10 = acc10[i] + bb0, v11 = acc11[i] + bb1;
    if (act_len) {
      if (!(t < act_len[mA] - 1)) { v00 = 0.0f; v01 = 0.0f; }
      if (!(t < act_len[mB] - 1)) { v10 = 0.0f; v11 = 0.0f; }
    }
    if (in0) C[(size_t)mA * ldc + na] = v00;
    if (in1) C[(size_t)mA * ldc + nb] = v01;
    if (in0) C[(size_t)mB * ldc + na] = v10;
    if (in1) C[(size_t)mB * ldc + nb] = v11;
  }
}

// ---------------------------------------------------------------------------
// mean over pixels: mean[b][e] = (1/P) sum_p enc[b][p][e]
// ---------------------------------------------------------------------------
__global__ __launch_bounds__(256) void mean_kernel(const float* __restrict__ enc,
                                                   float* __restrict__ mean) {
  int b = blockIdx.x;
  for (int e = threadIdx.x; e < NE; e += 256) {
    float s = 0.0f;
    for (int p = 0; p < NP; ++p) s += enc[((size_t)b * NP + p) * NE + e];
    mean[(size_t)b * NE + e] = s * (1.0f / (float)NP);
  }
}

// ---------------------------------------------------------------------------
// 32x32 LDS transpose: out[c][r] = in[r][c]; in is rows x cols. block (32,8)
// ---------------------------------------------------------------------------
__global__ __launch_bounds__(256) void transpose_kernel(const float* __restrict__ in,
                                                        float* __restrict__ out,
                                                        int rows, int cols) {
  __shared__ float tile[32][33];
  int c0 = blockIdx.x * 32, r0 = blockIdx.y * 32;
  int x = threadIdx.x, y = threadIdx.y;
#pragma unroll
  for (int i = 0; i < 32; i += 8)
    if (r0 + y + i < rows && c0 + x < cols)
      tile[y + i][x] = in[(size_t)(r0 + y + i) * cols + (c0 + x)];
  __syncthreads();
#pragma unroll
  for (int i = 0; i < 32; i += 8)
    if (c0 + y + i < cols && r0 + x < rows)
      out[(size_t)(c0 + y + i) * rows + (r0 + x)] = tile[x][y + i];
}

// ---------------------------------------------------------------------------
// Fused attention per batch element (one block of 256 threads per b):
//   e[p]     = sum_a relu(att1[b,p,a] + att2[b,a]) * w_full[a] + b_full
//   alpha    = softmax_p(e)
//   awe[b,j] = sum_p alpha[p] * enc[b,p,j]
//   alphas_out[b,t,p] = active ? alpha[p] : 0
// ---------------------------------------------------------------------------
__global__ __launch_bounds__(256) void att_kernel(
    const float* __restrict__ att1, const float* __restrict__ att2,
    const float* __restrict__ wfull, const float* __restrict__ bfull,
    const float* __restrict__ enc, float* __restrict__ alpha_out,
    float* __restrict__ awe, const int* __restrict__ len, int t) {
  __shared__ float att2_s[ND];
  __shared__ float wf_s[ND];
  __shared__ float e_s[NP];
  __shared__ float red_s[1];
  const int b = blockIdx.x;
  const int tid = threadIdx.x;

  for (int j = tid; j < ND; j += 256) {
    att2_s[j] = att2[(size_t)b * ND + j];
    wf_s[j] = wfull[j];
  }
  __syncthreads();

  const int wave = tid >> 5, lane = tid & 31;
  for (int p = wave; p < NP; p += 8) {
    const float* row = att1 + ((size_t)b * NP + p) * ND;
    float s = 0.0f;
    for (int a = lane; a < ND; a += 32) {
      float v = row[a] + att2_s[a];
      s += fmaxf(v, 0.0f) * wf_s[a];
    }
#pragma unroll
    for (int off = 16; off > 0; off >>= 1) s += __shfl_xor(s, off, 32);
    if (lane == 0) e_s[p] = s + bfull[0];
  }
  __syncthreads();

  if (tid == 0) {
    float mx = -3.0e38f;
    for (int p = 0; p < NP; ++p) mx = fmaxf(mx, e_s[p]);
    float sum = 0.0f;
    for (int p = 0; p < NP; ++p) {
      float v = expf(e_s[p] - mx);
      e_s[p] = v;
      sum += v;
    }
    red_s[0] = 1.0f / sum;
  }
  __syncthreads();

  const float inv = red_s[0];
  const bool active = t < (len[b] - 1);
  if (tid < NP) {
    float a = e_s[tid] * inv;
    e_s[tid] = a;
    alpha_out[(size_t)b * (NT * NP) + tid] = active ? a : 0.0f;
  }
  __syncthreads();

  for (int j = tid; j < NE; j += 256) {
    float acc = 0.0f;
    for (int p = 0; p < NP; ++p)
      acc += e_s[p] * enc[((size_t)b * NP + p) * NE + j];
    awe[(size_t)b * NE + j] = acc;
  }
}

// ---------------------------------------------------------------------------
// x = concat(emb_table[token], sigmoid(fpre) * awe)   — one block per batch b
// ---------------------------------------------------------------------------
__global__ __launch_bounds__(256) void xbuild_kernel(
    const float* __restrict__ emb_table, const int* __restrict__ caps,
    const float* __restrict__ fpre, const float* __restrict__ awe,
    float* __restrict__ x, int t) {
  const int b = blockIdx.x;
  const int tok = caps[b * MAXLEN + t];
  for (int j = threadIdx.x; j < ND + NE; j += 256) {
    float v;
    if (j < ND) {
      v = emb_table[(size_t)tok * ND + j];
    } else {
      int jj = j - ND;
      v = sigmoidf_(fpre[(size_t)b * NE + jj]) * awe[(size_t)b * NE + jj];
    }
    x[(size_t)b * (ND + NE) + j] = v;
  }
}

// ---------------------------------------------------------------------------
// LSTM pointwise + masked state update. gates layout per b: [i | f | g | o]
// ---------------------------------------------------------------------------
__global__ __launch_bounds__(256) void lstm_kernel(const float* __restrict__ gates,
                                                   float* __restrict__ h,
                                                   float* __restrict__ c,
                                                   const int* __restrict__ len,
                                                   int t) {
  int idx = blockIdx.x * 256 + threadIdx.x;  // NB*ND total
  int b = idx >> 9, j = idx & (ND - 1);
  bool active = t < (len[b] - 1);
  const float* g = gates + (size_t)b * (4 * ND);
  float ig = g[j], fg = g[ND + j], gg = g[2 * ND + j], og = g[3 * ND + j];
  float cn = sigmoidf_(fg) * c[idx] + sigmoidf_(ig) * tanhf(gg);
  float hn = sigmoidf_(og) * tanhf(cn);
  if (active) {
    c[idx] = cn;
    h[idx] = hn;
  }
}

// ---------------------------------------------------------------------------
// Host side
// ---------------------------------------------------------------------------
static inline void gemm(hipStream_t s, const float* A, int lda, const float* B,
                        int ldb, const float* bias1, const float* bias2, float* C,
                        int ldc, int M, int N, int K, const float* A2 = nullptr,
                        int lda2 = 0, const float* B2 = nullptr, int ldb2 = 0,
                        int K2 = 0, const int* act_len = nullptr, int t = 0) {
  dim3 grid((N + 127) / 128, M / 64);
  gemm_wmma_f32<<<grid, dim3(256), 0, s>>>(A, lda, B, ldb, A2, lda2, B2, ldb2,
                                           bias1, bias2, C, ldc, N, K, K2,
                                           act_len, t);
}

extern "C" void kernel_launch(void* const* d_in, const int* in_sizes, int n_in,
                              void* d_out, int out_size, void* d_ws, size_t ws_size,
                              hipStream_t stream) {
  const float* enc       = (const float*)d_in[0];
  const int*   caps      = (const int*)d_in[1];
  const int*   lens      = (const int*)d_in[2];
  const float* emb_table = (const float*)d_in[3];
  const float* W_enc_att = (const float*)d_in[4];
  const float* b_enc_att = (const float*)d_in[5];
  const float* W_dec_att = (const float*)d_in[6];
  const float* b_dec_att = (const float*)d_in[7];
  const float* w_full    = (const float*)d_in[8];
  const float* b_full    = (const float*)d_in[9];
  const float* W_init_h  = (const float*)d_in[10];
  const float* b_init_h  = (const float*)d_in[11];
  const float* W_init_c  = (const float*)d_in[12];
  const float* b_init_c  = (const float*)d_in[13];
  const float* W_fbeta   = (const float*)d_in[14];
  const float* b_fbeta   = (const float*)d_in[15];
  const float* W_ih      = (const float*)d_in[16];
  const float* b_ih      = (const float*)d_in[17];
  const float* W_hh      = (const float*)d_in[18];
  const float* b_hh      = (const float*)d_in[19];
  const float* W_fc      = (const float*)d_in[20];
  const float* b_fc      = (const float*)d_in[21];

  float* preds_out  = (float*)d_out;                            // (B, T, V)
  float* alphas_out = preds_out + (size_t)NB * NT * NV;         // (B, T, P)

  // Workspace carve-up (floats)
  float* ws = (float*)d_ws;
  size_t off = 0;
  float* mean  = ws + off; off += (size_t)NB * NE;            // 128x2048
  float* h     = ws + off; off += (size_t)NB * ND;            // 128x512
  float* c     = ws + off; off += (size_t)NB * ND;            // 128x512
  float* att1  = ws + off; off += (size_t)NB * NP * ND;       // 25088x512
  float* att2  = ws + off; off += (size_t)NB * ND;            // 128x512
  float* awe   = ws + off; off += (size_t)NB * NE;            // 128x2048
  float* fpre  = ws + off; off += (size_t)NB * NE;            // 128x2048
  float* x     = ws + off; off += (size_t)NB * (ND + NE);     // 128x2560
  float* gates = ws + off; off += (size_t)NB * 4 * ND;        // 128x2048
  float* W_ihT = ws + off; off += (size_t)(ND + NE) * 4 * ND; // 2560x2048
  float* W_hhT = ws + off; off += (size_t)ND * 4 * ND;        // 512x2048

  // --- One-time prep --------------------------------------------------------
  mean_kernel<<<NB, 256, 0, stream>>>(enc, mean);
  // W_ih is (2048, 2560) -> W_ihT (2560, 2048)
  transpose_kernel<<<dim3((ND + NE) / 32, (4 * ND) / 32), dim3(32, 8), 0, stream>>>(
      W_ih, W_ihT, 4 * ND, ND + NE);
  // W_hh is (2048, 512) -> W_hhT (512, 2048)
  transpose_kernel<<<dim3(ND / 32, (4 * ND) / 32), dim3(32, 8), 0, stream>>>(
      W_hh, W_hhT, 4 * ND, ND);
  // h0 / c0 = mean @ W_init_{h,c} + b
  gemm(stream, mean, NE, W_init_h, ND, b_init_h, nullptr, h, ND, NB, ND, NE);
  gemm(stream, mean, NE, W_init_c, ND, b_init_c, nullptr, c, ND, NB, ND, NE);
  // att1 = enc @ W_enc_att + b  (25088 x 512, K = 2048)
  gemm(stream, enc, NE, W_enc_att, ND, b_enc_att, nullptr, att1, ND, NB * NP, ND, NE);

  // --- Sequential decode ----------------------------------------------------
  for (int t = 0; t < NT; ++t) {
    // att2 = h @ W_dec_att + b
    gemm(stream, h, ND, W_dec_att, ND, b_dec_att, nullptr, att2, ND, NB, ND, ND);
    // attention: e -> softmax -> alpha (masked out) -> awe
    att_kernel<<<NB, 256, 0, stream>>>(att1, att2, w_full, b_full, enc,
                                       alphas_out + (size_t)t * NP, awe, lens, t);
    // fpre = h @ W_fbeta + b  (gate pre-activation)
    gemm(stream, h, ND, W_fbeta, NE, b_fbeta, nullptr, fpre, NE, NB, NE, ND);
    // x = [emb_t , sigmoid(fpre) * awe]
    xbuild_kernel<<<NB, 256, 0, stream>>>(emb_table, caps, fpre, awe, x, t);
    // gates = x @ W_ih.T + b_ih + h @ W_hh.T + b_hh
    gemm(stream, x, ND + NE, W_ihT, 4 * ND, b_ih, b_hh, gates, 4 * ND, NB, 4 * ND,
         ND + NE, h, ND, W_hhT, 4 * ND, ND);
    // masked LSTM state update
    lstm_kernel<<<(NB * ND) / 256, 256, 0, stream>>>(gates, h, c, lens, t);
    // preds = h_new @ W_fc + b_fc, masked rows -> 0; strided into (B, T, V)
    gemm(stream, h, ND, W_fc, NV, b_fc, nullptr, preds_out + (size_t)t * NV,
         NT * NV, NB, NV, ND, nullptr, 0, nullptr, 0, 0, lens, t);
  }
}